// PairwiseScore_74208444941029
// MI455X (gfx1250) — compile-verified
//
#include <hip/hip_runtime.h>
#include <hip/hip_bf16.h>

// ---------------------------------------------------------------------------
// PairwiseScore fused kernel for gfx1250 (MI455X, wave32, WMMA)
//
//   Layer1 GEMM [P,1596]x[1596,150] with V_WMMA_F32_16X16X4_F32 (fp32
//   fidelity). Features are built on the fly into LDS (never hit HBM).
//   W1 chunk is stored in a paired-k-row interleaved LDS layout so each
//   B fragment is a single aligned ds_load_b64 (no v_mov fix-up).
// ---------------------------------------------------------------------------

namespace {
constexpr int S_SPANS = 4096;
constexpr int K_ANT   = 64;
constexpr int G_DIM   = 512;
constexpr int P_PAIR  = S_SPANS * K_ANT;   // 262144
constexpr int F_DIM   = 1600;              // 3*G + 60 padded to 1600
constexpr int H_DIM   = 150;
constexpr int H_PAD   = 160;               // 10 N-tiles of 16
constexpr int KC      = 32;                // feature (K) chunk
constexpr int N_CHUNK = F_DIM / KC;        // 50
constexpr int RSF     = 36;                // feature LDS row stride (bank-safe)
constexpr int RSW2    = 352;               // W1 LDS row-PAIR stride (352%64==32)
}

typedef __attribute__((ext_vector_type(2))) float v2f;
typedef __attribute__((ext_vector_type(8))) float v8f;

__global__ __launch_bounds__(256)
void pair_mlp_wmma_kernel(const float* __restrict__ g_i,
                          const float* __restrict__ mention_scores,
                          const float* __restrict__ dist_embed,
                          const float* __restrict__ genre_embed,
                          const float* __restrict__ speaker_embed,
                          const float* __restrict__ W1,
                          const float* __restrict__ b1,
                          const float* __restrict__ W2,
                          const float* __restrict__ b2,
                          const int*   __restrict__ mention_ids,
                          const int*   __restrict__ antecedent_ids,
                          const int*   __restrict__ distances,
                          const int*   __restrict__ genres,
                          const int*   __restrict__ speakers,
                          float*       __restrict__ out)
{
    __shared__ float featL[8][16 * RSF];     // per-wave 16x32 feature tile
    __shared__ float w1c[2][16 * RSW2];      // double-buffered 32x160 W1 tile,
                                             // paired-row interleaved layout:
                                             // (kk,n) -> (kk>>1)*RSW2 + 2n + (kk&1)
    __shared__ float red[8][16];             // per-wave row-score exchange

    const int tid   = threadIdx.x;
    const int w     = tid >> 5;              // wave id 0..7
    const int lane  = tid & 31;
    const int pbase = blockIdx.x * 128 + w * 16;   // first pair of this wave

    v8f acc[10];
    const v8f vzero = {0.f, 0.f, 0.f, 0.f, 0.f, 0.f, 0.f, 0.f};
#pragma unroll
    for (int nt = 0; nt < 10; ++nt) acc[nt] = vzero;

    const int row16 = lane & 15;             // A-matrix M row / B-matrix N col
    const int koff  = (lane >> 4) << 1;      // K offset 0 (lanes 0-15) / 2 (16-31)
    const int hi    = lane >> 4;             // row-pair select for B

    for (int c = 0; c < N_CHUNK; ++c) {
        float* buf = w1c[c & 1];

        // ---- stage W1 chunk [c*KC .. +31] x [0..159] (paired-row layout) ---
        {
            const int kbase = c * KC;
#pragma unroll
            for (int it = 0; it < 10; ++it) {          // 16*160 / 256 = 10
                const int i  = it * 256 + tid;
                const int kp = i / H_PAD;              // row pair 0..15
                const int n  = i % H_PAD;              // col 0..159
                const int r0 = kbase + 2 * kp;
                const int r1 = r0 + 1;
                float2 v;
                v.x = (r0 < 1596 && n < H_DIM) ? W1[(size_t)r0 * H_DIM + n] : 0.f;
                v.y = (r1 < 1596 && n < H_DIM) ? W1[(size_t)r1 * H_DIM + n] : 0.f;
                *(float2*)&buf[kp * RSW2 + 2 * n] = v;
            }
        }

        // ---- stage this wave's 16x32 feature tile into its own LDS region --
        if (c < 48) {
            const int region  = c >> 4;                // 0:i_g 1:j_g 2:i*j
            const int colbase = (c & 15) * 32;
#pragma unroll
            for (int pass = 0; pass < 4; ++pass) {
                const int row = pass * 4 + (lane >> 3);
                const int cl  = (lane & 7) * 4;
                const int p   = pbase + row;
                float4 val;
                if (region == 0) {
                    const int mid = mention_ids[p];
                    val = *(const float4*)&g_i[(size_t)mid * G_DIM + colbase + cl];
                } else if (region == 1) {
                    const int aid = antecedent_ids[p];
                    val = *(const float4*)&g_i[(size_t)aid * G_DIM + colbase + cl];
                } else {
                    const int mid = mention_ids[p];
                    const int aid = antecedent_ids[p];
                    const float4 a4 = *(const float4*)&g_i[(size_t)mid * G_DIM + colbase + cl];
                    const float4 b4 = *(const float4*)&g_i[(size_t)aid * G_DIM + colbase + cl];
                    val.x = a4.x * b4.x; val.y = a4.y * b4.y;
                    val.z = a4.z * b4.z; val.w = a4.w * b4.w;
                }
                *(float4*)&featL[w][row * RSF + cl] = val;
            }
        } else {
            // phi embeddings: [dist(20) | genre(20) | speaker(20) | pad(4)]
            const int pcbase = (c - 48) * 32;
#pragma unroll
            for (int pass = 0; pass < 4; ++pass) {
                const int row = pass * 4 + (lane >> 3);
                const int cl  = (lane & 7) * 4;
                const int p   = pbase + row;
                const int d   = distances[p];
                const int gg  = genres[p];
                const int sp  = speakers[p];
                const int bucket = (d >= 1) + (d >= 2) + (d >= 3) + (d >= 4) +
                                   (d >= 8) + (d >= 16) + (d >= 32) + (d >= 64);
                float vv[4];
#pragma unroll
                for (int ii = 0; ii < 4; ++ii) {
                    const int pc = pcbase + cl + ii;
                    float x = 0.f;
                    if (pc < 20)      x = dist_embed[bucket * 20 + pc];
                    else if (pc < 40) x = genre_embed[gg * 20 + (pc - 20)];
                    else if (pc < 60) x = speaker_embed[sp * 20 + (pc - 40)];
                    vv[ii] = x;
                }
                float4 val; val.x = vv[0]; val.y = vv[1]; val.z = vv[2]; val.w = vv[3];
                *(float4*)&featL[w][row * RSF + cl] = val;
            }
        }

        __syncthreads();   // single barrier per chunk (W1 double-buffered)

        // ---- 8 k-steps x 10 N-tiles of fp32 WMMA ---------------------------
        const float* fptr = &featL[w][row16 * RSF + koff];
#pragma unroll
        for (int ks = 0; ks < 8; ++ks) {
            const v2f a = *(const v2f*)(fptr + ks * 4);
            const float* bb = &buf[(2 * ks + hi) * RSW2 + 2 * row16];
#pragma unroll
            for (int nt = 0; nt < 10; ++nt) {
                const v2f b = *(const v2f*)(bb + nt * 32);
                acc[nt] = __builtin_amdgcn_wmma_f32_16x16x4_f32(
                    false, a, false, b, (short)0, acc[nt], false, false);
            }
        }
    }

    // ---- epilogue: bias + ReLU + W2 dot + score assembly -------------------
    float part[8] = {0.f, 0.f, 0.f, 0.f, 0.f, 0.f, 0.f, 0.f};
#pragma unroll
    for (int nt = 0; nt < 10; ++nt) {
        const int col = nt * 16 + row16;
        if (col < H_DIM) {
            const float w2v = W2[col];
            const float b1v = b1[col];
#pragma unroll
            for (int v = 0; v < 8; ++v) {
                float h = acc[nt][v] + b1v;
                h = h > 0.f ? h : 0.f;
                part[v] = fmaf(h, w2v, part[v]);
            }
        }
    }
    // reduce each vgpr slot across its 16-lane half (rows 0-7 / 8-15)
#pragma unroll
    for (int v = 0; v < 8; ++v) {
#pragma unroll
        for (int off = 1; off < 16; off <<= 1)
            part[v] += __shfl_xor(part[v], off, 32);
    }
    if ((lane & 15) == 0) {
        const int half = lane >> 4;
#pragma unroll
        for (int v = 0; v < 8; ++v) red[w][half * 8 + v] = part[v];
    }
    asm volatile("s_wait_dscnt 0x0" ::: "memory");  // same-wave LDS exchange
    if (lane < 16) {
        const float sij = red[w][lane];
        const int p   = pbase + lane;
        const int mid = mention_ids[p];
        const int aid = antecedent_ids[p];
        const float val = sij + b2[0] + mention_scores[mid] + mention_scores[aid];
        out[(p >> 6) * 65 + (p & 63)] = val;   // span-major [S, K+1] layout
    }
}

// ---------------------------------------------------------------------------
// Masked softmax over K=64 antecedents + epsilon column; pad invalid = 1000.
// One wave32 per span.
// ---------------------------------------------------------------------------
__global__ __launch_bounds__(256)
void span_softmax_kernel(const int* __restrict__ ant_counts,
                         float* __restrict__ out)
{
    const int w    = threadIdx.x >> 5;
    const int lane = threadIdx.x & 31;
    const int s    = blockIdx.x * 8 + w;

    const int cnt = ant_counts[s];
    float* rowp = out + (size_t)s * 65;

    const float l0 = rowp[lane];
    const float l1 = rowp[lane + 32];
    const bool v0 = lane < cnt;
    const bool v1 = (lane + 32) < cnt;
    const float m0 = v0 ? l0 : -1e9f;
    const float m1 = v1 ? l1 : -1e9f;

    float m = fmaxf(fmaxf(m0, m1), 0.f);      // epsilon logit = 0
#pragma unroll
    for (int off = 16; off >= 1; off >>= 1)
        m = fmaxf(m, __shfl_xor(m, off, 32));

    const float e0 = v0 ? expf(m0 - m) : 0.f;
    const float e1 = v1 ? expf(m1 - m) : 0.f;
    float esum = e0 + e1;
#pragma unroll
    for (int off = 16; off >= 1; off >>= 1)
        esum += __shfl_xor(esum, off, 32);

    const float ee  = expf(0.f - m);
    const float inv = 1.f / (esum + ee);

    rowp[lane]      = v0 ? e0 * inv : 1000.0f;
    rowp[lane + 32] = v1 ? e1 * inv : 1000.0f;
    if (lane == 0) rowp[64] = ee * inv;
}

extern "C" void kernel_launch(void* const* d_in, const int* in_sizes, int n_in,
                              void* d_out, int out_size, void* d_ws, size_t ws_size,
                              hipStream_t stream) {
    (void)in_sizes; (void)n_in; (void)d_ws; (void)ws_size; (void)out_size;

    const float* g_i            = (const float*)d_in[0];
    const float* mention_scores = (const float*)d_in[1];
    const float* dist_embed     = (const float*)d_in[2];
    const float* genre_embed    = (const float*)d_in[3];
    const float* speaker_embed  = (const float*)d_in[4];
    const float* W1             = (const float*)d_in[5];
    const float* b1             = (const float*)d_in[6];
    const float* W2             = (const float*)d_in[7];
    const float* b2             = (const float*)d_in[8];
    const int*   mention_ids    = (const int*)d_in[9];
    const int*   antecedent_ids = (const int*)d_in[10];
    const int*   distances      = (const int*)d_in[11];
    const int*   genres         = (const int*)d_in[12];
    const int*   speakers       = (const int*)d_in[13];
    const int*   ant_counts     = (const int*)d_in[14];
    float* out = (float*)d_out;

    // 262144 pairs / 128 per block = 2048 blocks of 8 waves
    pair_mlp_wmma_kernel<<<P_PAIR / 128, 256, 0, stream>>>(
        g_i, mention_scores, dist_embed, genre_embed, speaker_embed,
        W1, b1, W2, b2, mention_ids, antecedent_ids,
        distances, genres, speakers, out);

    // 4096 spans / 8 waves per block = 512 blocks
    span_softmax_kernel<<<S_SPANS / 8, 256, 0, stream>>>(ant_counts, out);
}